// LwOpenPose_28424093565189
// MI455X (gfx1250) — compile-verified
//
#include <hip/hip_runtime.h>
#include <hip/hip_bf16.h>

typedef __attribute__((ext_vector_type(2))) float v2f;
typedef __attribute__((ext_vector_type(8))) float v8f;

#define HSZ 368
#define WSZ 368
#define PLANE (HSZ * WSZ)
#define KC 64
#define NCHAN (8 * 19)          // 152 (b, joint) channels
#define BIGKEY 0x7FFFFFFF

// d_out layout (floats, concatenated in reference return order):
//   conn   [8,19,64,64]  at 0          (622592)
//   coords [8,19,64,2]   at 622592     (19456)
//   score  [8,19,64]     at 642048     (9728)
//   valid  [8,19,64]     at 651776     (9728)
#define CONN_N   622592
#define COORD_OFF 622592
#define SCORE_OFF 642048
#define VALID_OFF 651776

__device__ __forceinline__ float thr01(float v) { return v < 0.1f ? 0.0f : v; }

// ---------------------------------------------------------------------------
// Kernel 1: one workgroup (512 threads = 16 waves) per (batch, joint) channel.
// Peak detect -> column histogram -> cutoff column -> collect -> bitonic sort
// -> WMMA pairwise-d2 -> greedy NMS -> write coords/score/valid.
// ---------------------------------------------------------------------------
__global__ __launch_bounds__(512) void peaks_kernel(const float* __restrict__ hm,
                                                    float* __restrict__ out)
{
    const int chan = blockIdx.x;                 // b*19 + joint
    const float* plane = hm + (size_t)chan * PLANE;
    const int tid = threadIdx.x;

    __shared__ int   colcnt[WSZ];
    __shared__ int   keys[256];
    __shared__ int   s_cstop, s_cnt;
    __shared__ float xs[KC], ys[KC], sc[KC];
    __shared__ int   vld[KC], keep[KC], supp[KC];
    __shared__ unsigned char closem[KC * KC];

    for (int i = tid; i < WSZ; i += 512) colcnt[i] = 0;
    if (tid == 0) s_cnt = 0;
    __syncthreads();

    // ---- pass A: strict 4-neighbor local max on thresholded map; count per column
    for (int p = tid; p < PLANE; p += 512) {
        int r = p / WSZ, c = p - r * WSZ;
        float v  = thr01(plane[p]);
        float nr = (c + 1 < WSZ) ? thr01(plane[p + 1])   : 0.0f;
        float nl = (c     > 0  ) ? thr01(plane[p - 1])   : 0.0f;
        float nd = (r + 1 < HSZ) ? thr01(plane[p + WSZ]) : 0.0f;
        float nu = (r     > 0  ) ? thr01(plane[p - WSZ]) : 0.0f;
        if (v > nr && v > nl && v > nd && v > nu) atomicAdd(&colcnt[c], 1);
    }
    __syncthreads();

    if (tid == 0) {
        int cum = 0, cstop = WSZ - 1;
        for (int c = 0; c < WSZ; ++c) {
            cum += colcnt[c];
            if (cum >= KC) { cstop = c; break; }
        }
        s_cstop = cstop;
    }
    __syncthreads();
    const int cstop = s_cstop;

    // ---- pass B: re-scan columns [0, cstop]; collect keys (key = x*H + y).
    // Bound: <= 63 + 184 (no two vertically adjacent strict maxima) <= 247 < 256.
    {
        const int ncols = cstop + 1;
        const int tot = HSZ * ncols;
        for (int q = tid; q < tot; q += 512) {
            int r = q / ncols, c = q - r * ncols;
            int p = r * WSZ + c;
            float v  = thr01(plane[p]);
            float nr = (c + 1 < WSZ) ? thr01(plane[p + 1])   : 0.0f;
            float nl = (c     > 0  ) ? thr01(plane[p - 1])   : 0.0f;
            float nd = (r + 1 < HSZ) ? thr01(plane[p + WSZ]) : 0.0f;
            float nu = (r     > 0  ) ? thr01(plane[p - WSZ]) : 0.0f;
            if (v > nr && v > nl && v > nd && v > nu) {
                int idx = atomicAdd(&s_cnt, 1);
                if (idx < 256) keys[idx] = c * HSZ + r;
            }
        }
    }
    __syncthreads();
    {
        int cnt = s_cnt; if (cnt > 256) cnt = 256;
        if (tid < 256 && tid >= cnt) keys[tid] = BIGKEY;
    }

    // ---- bitonic sort 256 keys ascending (first 256 threads own one slot each)
    for (int kk = 2; kk <= 256; kk <<= 1) {
        for (int jj = kk >> 1; jj > 0; jj >>= 1) {
            __syncthreads();
            if (tid < 256) {
                int ixj = tid ^ jj;
                if (ixj > tid) {
                    int a = keys[tid], b = keys[ixj];
                    bool up = (tid & kk) == 0;
                    if ((a > b) == up) { keys[tid] = b; keys[ixj] = a; }
                }
            }
        }
    }
    __syncthreads();

    // ---- extract 64 candidates
    if (tid < KC) {
        int key = keys[tid];
        int isv = (key != BIGKEY);
        int c = key / HSZ, r = key - c * HSZ;
        if (!isv) { c = 0; r = 0; }
        xs[tid] = (float)c;
        ys[tid] = (float)r;
        sc[tid] = thr01(plane[r * WSZ + c]);
        vld[tid] = isv;
        supp[tid] = 0;
    }
    __syncthreads();

    // ---- pairwise squared distances via V_WMMA_F32_16X16X4_F32.
    // d2[m,n] = [x_m, y_m, |p_m|^2, 1] . [-2x_n, -2y_n, 1, |p_n|^2]  (K=4, C=0)
    // All operands are small integers -> exact in f32 in any accumulation order.
    {
        const int lane = tid & 31;
        const int wave = tid >> 5;          // 16 waves -> 16 (ti,tj) tiles
        const int ti = wave >> 2, tj = wave & 3;
        const int nloc = lane & 15;
        const int hi = lane >> 4;           // lane half selects K-pair
        const int m = ti * 16 + nloc;
        const int n = tj * 16 + nloc;
        float xm = xs[m], ym = ys[m], xn = xs[n], yn = ys[n];
        v2f a, b;
        a.x = hi ? (xm * xm + ym * ym) : xm;      // A: K=0/2 in vgpr0
        a.y = hi ? 1.0f                : ym;      //    K=1/3 in vgpr1
        b.x = hi ? 1.0f                : (-2.0f * xn);
        b.y = hi ? (xn * xn + yn * yn) : (-2.0f * yn);
        v8f cz = {};
        v8f d = __builtin_amdgcn_wmma_f32_16x16x4_f32(
            false, a, false, b, (short)0, cz, false, false);
#pragma unroll
        for (int v = 0; v < 8; ++v) {
            int mm = ti * 16 + v + 8 * hi;
            closem[mm * KC + tj * 16 + nloc] = (d[v] < 36.0f) ? 1 : 0;
        }
    }
    __syncthreads();

    // ---- greedy NMS (serial dependency; candidates sorted by key)
    if (tid == 0) {
        for (int i = 0; i < KC; ++i) {
            int ki = vld[i] && !supp[i];
            keep[i] = ki;
            if (ki) {
                const unsigned char* row = &closem[i * KC];
                for (int j = i + 1; j < KC; ++j)
                    if (row[j]) supp[j] = 1;
            }
        }
    }
    __syncthreads();

    // ---- write coords (2x), score, post-NMS valid
    if (tid < KC) {
        int base = chan * KC + tid;
        out[COORD_OFF + base * 2 + 0] = xs[tid] * 2.0f;
        out[COORD_OFF + base * 2 + 1] = ys[tid] * 2.0f;
        out[SCORE_OFF + base] = sc[tid];
        out[VALID_OFF + base] = keep[tid] ? 1.0f : 0.0f;
    }
}

// ---------------------------------------------------------------------------
// Kernel 2: PAF line-integral scoring. grid (152, 4): block handles 1024 of the
// 4096 (i,j) pairs for one (batch, limb). Reads candidate state from d_out.
// ---------------------------------------------------------------------------
__device__ __constant__ int cBODY_A[19] = {1,1,2,3,5,6,1,8,9,1,11,12,1,0,14,0,15,2,5};
__device__ __constant__ int cBODY_B[19] = {2,5,3,4,6,7,8,9,10,11,12,13,0,14,16,15,17,16,17};
__device__ __constant__ int cPAF0[19]   = {12,20,14,16,22,24,0,2,4,6,8,10,28,30,34,32,36,18,26};

__global__ __launch_bounds__(256) void paf_kernel(const float* __restrict__ paf,
                                                  float* out)
{
    const int bw = blockIdx.x;              // b*19 + limb
    const int b = bw / 19, l = bw - b * 19;
    const int tid = threadIdx.x;

    const int ja = cBODY_A[l], jb = cBODY_B[l];
    const int c0 = cPAF0[l];
    const float* pafx = paf + ((size_t)b * 38 + c0) * PLANE;
    const float* pafy = pafx + PLANE;

    __shared__ float axs[KC], ays[KC], sas[KC];
    __shared__ float bxs[KC], bys[KC], sbs[KC];
    __shared__ int   vas[KC], vbs[KC];

    if (tid < KC) {
        int ia = (b * 19 + ja) * KC + tid;
        axs[tid] = out[COORD_OFF + ia * 2 + 0] * 0.5f;
        ays[tid] = out[COORD_OFF + ia * 2 + 1] * 0.5f;
        sas[tid] = out[SCORE_OFF + ia];
        vas[tid] = (out[VALID_OFF + ia] != 0.0f);
    } else if (tid < 2 * KC) {
        int k = tid - KC;
        int ib = (b * 19 + jb) * KC + k;
        bxs[k] = out[COORD_OFF + ib * 2 + 0] * 0.5f;
        bys[k] = out[COORD_OFF + ib * 2 + 1] * 0.5f;
        sbs[k] = out[SCORE_OFF + ib];
        vbs[k] = (out[VALID_OFF + ib] != 0.0f);
    }
    __syncthreads();

    const int pair0 = blockIdx.y * 1024;
    float* conn = out + (size_t)bw * (KC * KC);

    for (int pp = pair0 + tid; pp < pair0 + 1024; pp += 256) {
        int i = pp >> 6, j = pp & 63;
        float ax = axs[i], ay = ays[i];
        float vx = bxs[j] - ax, vy = bys[j] - ay;
        float norm = sqrtf(vx * vx + vy * vy);
        float nrm = fmaxf(norm, 1e-6f);
        float ux = vx / nrm, uy = vy / nrm;

        int npass = 0;
        float sum = 0.0f;
#pragma unroll
        for (int p = 0; p < 10; ++p) {
            float t = (float)p / 9.0f;                 // folds to f32 constants
            int px = (int)rintf(ax + vx * t);          // round half-to-even == jnp.round
            int py = (int)rintf(ay + vy * t);
            px = px < 0 ? 0 : (px > WSZ - 1 ? WSZ - 1 : px);
            py = py < 0 ? 0 : (py > HSZ - 1 ? HSZ - 1 : py);
            int off = py * WSZ + px;
            float ps = ux * pafx[off] + uy * pafy[off];
            if (ps > 0.05f) { ++npass; sum += ps; }
        }
        float ratio = sum / (float)(npass > 0 ? npass : 1)
                    + fminf(184.0f / nrm - 1.0f, 0.0f);
        bool ok = (norm > 0.0f) && (ratio > 0.0f) && (npass >= 9)
                  && vas[i] && vbs[j];
        conn[pp] = ok ? (ratio + sas[i] + sbs[j]) : 0.0f;
    }
}

// ---------------------------------------------------------------------------
extern "C" void kernel_launch(void* const* d_in, const int* in_sizes, int n_in,
                              void* d_out, int out_size, void* d_ws, size_t ws_size,
                              hipStream_t stream) {
    (void)in_sizes; (void)n_in; (void)out_size; (void)d_ws; (void)ws_size;
    const float* hm  = (const float*)d_in[0];   // [8,19,368,368]
    const float* paf = (const float*)d_in[1];   // [8,38,368,368]
    float* out = (float*)d_out;

    peaks_kernel<<<dim3(NCHAN), dim3(512), 0, stream>>>(hm, out);
    paf_kernel<<<dim3(NCHAN, 4), dim3(256), 0, stream>>>(paf, out);
}